// ReservoirLayer_72206990180649
// MI455X (gfx1250) — compile-verified
//
#include <hip/hip_runtime.h>
#include <math.h>

typedef float v2f __attribute__((ext_vector_type(2)));
typedef float v8f __attribute__((ext_vector_type(8)));
typedef unsigned int v4u __attribute__((ext_vector_type(4)));
typedef int v4i __attribute__((ext_vector_type(4)));
typedef int v8i __attribute__((ext_vector_type(8)));

static constexpr int kB = 8;
static constexpr int kT = 4096;
static constexpr int kR = 1024;
static constexpr int kF = 64;
static constexpr int kI = 32;
static constexpr int kK = kF + kI;       // 96
static constexpr float kAlpha = 0.9f;    // LEAK_RATE

static constexpr int NBLK2 = 16;         // workgroups in the scan "cluster"
static constexpr int WS = 1028;          // W_hh LDS row stride (floats), %64==4 -> conflict-free B frags
static constexpr int HROW = 1032;        // h LDS effective row stride (TDM pad: +2 dw per 256 dw)
static constexpr int HWORDS = kB * HROW; // h LDS footprint in floats

#if __has_builtin(__builtin_amdgcn_tensor_load_to_lds) && __has_builtin(__builtin_amdgcn_s_wait_tensorcnt)
#define RL_USE_TDM 1
#endif

// h element (row, k) location in LDS, matching TDM padding: 2 dwords inserted
// after every 256 dwords streamed (row length 1024 -> row stride 1032).
__device__ __forceinline__ int hofs(int row, int k) {
    return row * HROW + k + ((k >> 8) << 1);
}

// ---------------------------------------------------------------------------
// init: zero barrier words + h ping-pong buffers in workspace
// ws layout (floats): [0..63] ctrl, [64 .. 64+8192) h0, [.. +8192) h1
// ---------------------------------------------------------------------------
__global__ void rl_init(float* __restrict__ wsf) {
    int i = blockIdx.x * blockDim.x + threadIdx.x;
    if (i < 64 + 2 * kB * kR) wsf[i] = 0.0f;
}

// ---------------------------------------------------------------------------
// Phase 1: drive[m, r] = X[m,:] @ W^T + b, X = [feedback | driving], K = 96.
// One wave per 16x16 output tile, V_WMMA_F32_16X16X4_F32, 24 WMMAs per tile.
// Stored into d_out (overwritten with states by phase 2).
// ---------------------------------------------------------------------------
__global__ __launch_bounds__(128) void rl_drive(
    const float* __restrict__ fb, const float* __restrict__ dr,
    const float* __restrict__ Wfb, const float* __restrict__ Win,
    const float* __restrict__ bias, float* __restrict__ out)
{
    const int lane = threadIdx.x & 31;
    const int wave = threadIdx.x >> 5;
    const int hf   = lane >> 4;          // wave half: K pair (A/B) or +8 rows (C/D)
    const int n16  = lane & 15;

    const int rowbase = blockIdx.x * 16;                      // 2048 row tiles
    const int col     = (blockIdx.y * 4 + wave) * 16 + n16;   // 16*4 waves cover R=1024

    const int m = rowbase + n16;                              // A-matrix row for this lane

    v8f c;
    const float bv = bias[col];
    #pragma unroll
    for (int v = 0; v < 8; ++v) c[v] = bv;

    #pragma unroll
    for (int k0 = 0; k0 < kK; k0 += 4) {
        v2f a, b;
        if (k0 < kF) {                  // compile-time after unroll
            const int k = k0 + hf * 2;
            a = *(const v2f*)(fb  + (size_t)m   * kF + k);
            b = *(const v2f*)(Wfb + (size_t)col * kF + k);
        } else {
            const int k = k0 - kF + hf * 2;
            a = *(const v2f*)(dr  + (size_t)m   * kI + k);
            b = *(const v2f*)(Win + (size_t)col * kI + k);
        }
        c = __builtin_amdgcn_wmma_f32_16x16x4_f32(false, a, false, b, (short)0, c, false, false);
    }

    #pragma unroll
    for (int v = 0; v < 8; ++v) {
        const int mrow = rowbase + v + hf * 8;                // C/D: VGPR v -> M=v / v+8
        out[(size_t)mrow * kR + col] = c[v];
    }
}

// ---------------------------------------------------------------------------
// Phase 2: sequential reservoir scan. 16 workgroups, each owns 64 rows of
// W_hh pinned in LDS. Per step: TDM-stage h (32 KB) into LDS, 4 waves each
// produce one 16x16 tile via 256 chained f32 WMMAs (K=1024), leak+tanh,
// store state + next-h, then device-wide barrier (+cluster barrier).
// ---------------------------------------------------------------------------
__global__ __launch_bounds__(128) void rl_scan(
    const float* __restrict__ Whh, float* out, float* wsf)
{
    extern __shared__ float lds[];
    float* Wlds = lds;                   // [64][WS]
    float* Hlds = lds + 64 * WS;         // TDM-padded [8][1024] (HWORDS floats)

    unsigned* bar_cnt = (unsigned*)wsf;
    unsigned* bar_gen = (unsigned*)wsf + 1;
    float* h0 = wsf + 64;
    float* h1 = h0 + kB * kR;

    const int tid  = threadIdx.x;
    const int lane = tid & 31;
    const int wave = tid >> 5;
    const int hf   = lane >> 4;
    const int n16  = lane & 15;
    const int wg   = blockIdx.x;                       // 0..15

    // ---- one-time: load this WG's 64 rows of W_hh into LDS ----
    {
        const float* src = Whh + (size_t)wg * 64 * kR;
        for (int q = tid; q < 64 * kR / 4; q += 128) {
            const int row = q >> 8;                    // 256 quads per row
            const int c4  = (q & 255) * 4;
            const float4 vv = *(const float4*)(src + (size_t)row * kR + c4);
            float* dst = Wlds + row * WS + c4;
            dst[0] = vv.x; dst[1] = vv.y; dst[2] = vv.z; dst[3] = vv.w;
        }
    }
    __syncthreads();

    const int   colg  = wg * 64 + wave * 16 + n16;     // global output column (R index)
    const int   tileW = wave * 16;                     // W row base within LDS slice
    const float amask = (n16 < 8) ? 1.0f : 0.0f;       // A rows 8..15 = batch padding
    const int   arow  = n16 & 7;                       // h row (lanes n16 / n16+8 broadcast)

    for (int t = 0; t < kT; ++t) {
        const float* hprev = (t & 1) ? h1 : h0;
        float*       hnext = (t & 1) ? h0 : h1;

        // ---- stage h_prev[8][1024] into LDS ----
#ifdef RL_USE_TDM
        if (wave == 0) {
            // Tensor DMA descriptor (D#): 2D tensor 1024x8, f32, LDS padding
            // 2 dwords / 256 dwords -> landed row stride = 1032 floats.
            const unsigned long long ga = (unsigned long long)(const void*)hprev;
            v4u g0;
            g0[0] = 1u;                                         // count=1 (user mode)
            g0[1] = (unsigned)(64 * WS) * 4u;                   // lds_addr (bytes) = Hlds
            g0[2] = (unsigned)(ga & 0xFFFFFFFFull);             // global_addr[31:0]
            g0[3] = (unsigned)((ga >> 32) & 0x01FFFFFFull)      // global_addr[56:32]
                  | (2u << 30);                                 // type = 2 ("image")
            v8i g1;
            g1[0] = (2 << 16)                                   // data_size = 4B
                  | (1 << 20)                                   // pad_enable
                  | (7 << 22)                                   // pad_interval = 256 dw
                  | (1 << 25);                                  // pad_amount = 2 dw
            g1[1] = (int)((unsigned)(kR & 0xFFFF) << 16);       // tensor_dim0[15:0] = 1024
            g1[2] = (int)((unsigned)kB << 16);                  // dim0[31:16]=0 | tensor_dim1 = 8
            g1[3] = (int)((unsigned)(kR & 0xFFFF) << 16);       // dim1[31:16]=0 | tile_dim0 = 1024
            g1[4] = kB;                                         // tile_dim1 = 8, tile_dim2 = 0
            g1[5] = kR;                                         // tensor_dim0_stride[31:0] = 1024
            g1[6] = 0;
            g1[7] = 0;
            v4i gz = {0, 0, 0, 0};
#if defined(__clang_major__) && __clang_major__ >= 23
            v8i gz8 = {0, 0, 0, 0, 0, 0, 0, 0};
            __builtin_amdgcn_tensor_load_to_lds(g0, g1, gz, gz, gz8, 0);
#else
            __builtin_amdgcn_tensor_load_to_lds(g0, g1, gz, gz, 0);
#endif
            __builtin_amdgcn_s_wait_tensorcnt(0);
        }
        __syncthreads();
#else
        for (int q = tid; q < kB * kR / 4; q += 128) {
            const int row = q >> 8;
            const int c4  = (q & 255) * 4;
            const float4 vv = *(const float4*)(hprev + (size_t)row * kR + c4);
            float* dst = Hlds + hofs(row, c4);
            dst[0] = vv.x; dst[1] = vv.y; dst[2] = vv.z; dst[3] = vv.w;
        }
        __syncthreads();
#endif

        // ---- C init from precomputed drive (phase 1 output, still in `out`) ----
        v8f c;
        #pragma unroll
        for (int v = 0; v < 8; ++v)
            c[v] = (hf == 0) ? out[((size_t)v * kT + t) * kR + colg] : 0.0f;

        if ((t + 1 < kT) && hf == 0)                    // global_prefetch_b8 next drive column
            __builtin_prefetch(&out[((size_t)(lane & 7) * kT + (t + 1)) * kR + colg], 0, 1);

        // ---- 16x16 tile of h @ W_hh^T, K = 1024 ----
        #pragma unroll 8
        for (int k0 = 0; k0 < kR; k0 += 4) {
            const int k = k0 + hf * 2;                  // lane's K pair within the 16x4 A tile
            v2f a = *(const v2f*)(Hlds + hofs(arow, k));
            a.x *= amask; a.y *= amask;                 // zero batch-pad rows 8..15
            const v2f b = *(const v2f*)(Wlds + (tileW + n16) * WS + k);
            c = __builtin_amdgcn_wmma_f32_16x16x4_f32(false, a, false, b, (short)0, c, false, false);
        }

        // ---- leak + tanh; write state (overwrites drive slot) + next h ----
        if (hf == 0) {
            #pragma unroll
            for (int v = 0; v < 8; ++v) {
                const float hp = Hlds[hofs(v, colg)];
                const float hn = tanhf((1.0f - kAlpha) * hp + kAlpha * c[v]);
                out[((size_t)v * kT + t) * kR + colg] = hn;
                hnext[(size_t)v * kR + colg] = hn;
            }
        }

        // ---- device-wide step barrier across the 16 resident blocks ----
        __threadfence();                                // release stores (device scope)
        __syncthreads();
        if (tid == 0) {
            const unsigned ticket = __hip_atomic_fetch_add(
                bar_cnt, 1u, __ATOMIC_ACQ_REL, __HIP_MEMORY_SCOPE_AGENT);
            if (ticket == (unsigned)(NBLK2 - 1)) {
                __hip_atomic_store(bar_cnt, 0u, __ATOMIC_RELAXED, __HIP_MEMORY_SCOPE_AGENT);
                __hip_atomic_fetch_add(bar_gen, 1u, __ATOMIC_RELEASE, __HIP_MEMORY_SCOPE_AGENT);
            } else {
                while (__hip_atomic_load(bar_gen, __ATOMIC_ACQUIRE,
                                         __HIP_MEMORY_SCOPE_AGENT) < (unsigned)(t + 1)) {
                    __builtin_amdgcn_s_sleep(1);
                }
            }
        }
        // cluster-hardware barrier: NOP when not dispatched as a cluster,
        // the fast s_barrier_signal/-wait(-3) path when it is.
        __builtin_amdgcn_s_cluster_barrier();
        __syncthreads();
        __threadfence();                                // acquire: no stale h in WGP$
    }
}

// ---------------------------------------------------------------------------
extern "C" void kernel_launch(void* const* d_in, const int* in_sizes, int n_in,
                              void* d_out, int out_size, void* d_ws, size_t ws_size,
                              hipStream_t stream) {
    const float* fb   = (const float*)d_in[0];   // feedback [8,4096,64]
    const float* dr   = (const float*)d_in[1];   // driving  [8,4096,32]
    const float* Wfb  = (const float*)d_in[2];   // [1024,64]
    const float* Win  = (const float*)d_in[3];   // [1024,32]
    const float* Whh  = (const float*)d_in[4];   // [1024,1024]
    const float* bias = (const float*)d_in[5];   // [1024]
    float* out = (float*)d_out;                  // [8,4096,1024]
    float* wsf = (float*)d_ws;

    const int initN = 64 + 2 * kB * kR;
    rl_init<<<(initN + 255) / 256, 256, 0, stream>>>(wsf);

    dim3 g1((kB * kT) / 16, kR / 64);            // (2048, 16) blocks, 4 waves each
    rl_drive<<<g1, 128, 0, stream>>>(fb, dr, Wfb, Win, bias, out);

    const size_t ldsBytes = (size_t)(64 * WS + HWORDS) * sizeof(float);  // 296,192 B
    rl_scan<<<NBLK2, 128, ldsBytes, stream>>>(Whh, out, wsf);
}